// SelectiveSSMKernel_36550171689111
// MI455X (gfx1250) — compile-verified
//
#include <hip/hip_runtime.h>

typedef __attribute__((ext_vector_type(2))) float v2f;
typedef __attribute__((ext_vector_type(4))) float v4f;
typedef __attribute__((ext_vector_type(8))) float v8f;

// ---------------------------------------------------------------------------
// Kernel 1: f32 WMMA GEMM.  Out_b = W (MxK, row-major) * U_b (KxL, row-major)
// Stored TRANSPOSED per batch: Ob[l*M + m]  (l-major, so the scan kernel's
// per-timestep reads are fully coalesced, and the 8 accumulator rows of one
// lane land on 8 consecutive floats -> two b128 stores).
// Block = 128 threads = 4 waves in a 2x2 grid of 16x16 WMMA tiles (32x32/blk).
// K, L are compile-time so all inner-loop addresses are immediate offsets.
// ---------------------------------------------------------------------------
template <int K, int L>
__global__ __launch_bounds__(128)
void gemm_f32_wmma(const float* __restrict__ W, const float* __restrict__ U,
                   float* __restrict__ Out, int M)
{
    const int b = blockIdx.z;
    const float* __restrict__ Ub = U + (long)b * K * L;
    float* __restrict__ Ob = Out + (long)b * (long)M * L;

    const int wave  = threadIdx.x >> 5;
    const int lane  = threadIdx.x & 31;
    const int m0    = blockIdx.y * 32 + (wave >> 1) * 16;
    const int l0    = blockIdx.x * 32 + (wave & 1) * 16;
    const int row   = lane & 15;   // M index for A, N index for B/C/D
    const int khalf = lane >> 4;   // selects K pair {0,1} vs {2,3}

    // Fold lane-dependent terms into base pointers; everything inside the
    // unrolled body is then a uniform immediate offset.
    const float* __restrict__ aptr = W  + (long)(m0 + row) * K + 2 * khalf;
    const float* __restrict__ bptr = Ub + (long)(2 * khalf) * L + l0 + row;

    v8f acc = {};
    for (int k = 0; k < K; k += 32) {
        #pragma unroll
        for (int j = 0; j < 8; ++j) {
            v2f a = *(const v2f*)(aptr + j * 4);        // A[m][kk], A[m][kk+1]
            v2f bm;
            bm.x = bptr[(long)(j * 4)     * L];         // B[kk  ][n]
            bm.y = bptr[(long)(j * 4 + 1) * L];         // B[kk+1][n]
            // 8 args: (neg_a, A, neg_b, B, c_mod, C, reuse_a, reuse_b)
            acc = __builtin_amdgcn_wmma_f32_16x16x4_f32(
                      false, a, false, bm, (short)0, acc, false, false);
        }
        aptr += 32;
        bptr += (long)32 * L;
    }

    // D layout: M = m0 + 8*khalf + r, N(l) = l0 + row.  Transposed store puts
    // the 8 r-values at consecutive addresses (32B aligned).
    float* __restrict__ dst = Ob + (long)(l0 + row) * M + m0 + 8 * khalf;
    v4f lo = {acc[0], acc[1], acc[2], acc[3]};
    v4f hi = {acc[4], acc[5], acc[6], acc[7]};
    *(v4f*)(dst)     = lo;
    *(v4f*)(dst + 4) = hi;
}

// ---------------------------------------------------------------------------
// Kernel 2: sequential selective scan over L.
// grid = (dmodel/64, batch), block = 128 threads.
// Thread t handles row d = d0 + t/2 and n-range [(t&1)*32, (t&1)*32+32);
// 32 h-state values live in VGPRs.  Per step: Bu/Cu column (64+64 f32,
// contiguous thanks to transposed layout) staged through LDS, then
// 64 muls + 64 FMAs per thread, 2-lane shuffle reduction for y.
// ---------------------------------------------------------------------------
__global__ __launch_bounds__(128)
void ssm_scan(const float* __restrict__ u, const float* __restrict__ A,
              const float* __restrict__ D,
              const float* __restrict__ BuT, const float* __restrict__ CuT,
              const float* __restrict__ dT, float* __restrict__ out,
              int dmodel, int N, int L)
{
    const int b  = blockIdx.y;
    const int t  = threadIdx.x;
    const int d  = blockIdx.x * 64 + (t >> 1);
    const int nh = (t & 1) * 32;

    const float* __restrict__ ubd = u   + ((size_t)b * dmodel + d) * L;
    float*       __restrict__ ybd = out + ((size_t)b * dmodel + d) * L;
    const float* __restrict__ Bub = BuT + (size_t)b * L * N;
    const float* __restrict__ Cub = CuT + (size_t)b * L * N;
    const float* __restrict__ dTb = dT  + (size_t)b * L * dmodel;

    const float Dd = D[d];

    float areg[32], h[32];
    #pragma unroll
    for (int i = 0; i < 32; ++i) { areg[i] = A[nh + i]; h[i] = 0.0f; }

    __shared__ float sB[64];
    __shared__ float sC[64];

    for (int l = 0; l < L; ++l) {
        if (t < 64) sB[t]      = Bub[(size_t)l * N + t];
        else        sC[t - 64] = Cub[(size_t)l * N + (t - 64)];
        __syncthreads();

        const float dl = dTb[(size_t)l * dmodel + d];
        const float ul = ubd[l];
        if (l + 1 < L) __builtin_prefetch(&ubd[l + 1], 0, 0);  // global_prefetch_b8
        const float du = dl * ul;

        float y = 0.0f;
        #pragma unroll
        for (int i = 0; i < 32; ++i) {
            const float hn = fmaf(dl * areg[i], h[i], du * sB[nh + i]);
            h[i] = hn;
            y = fmaf(sC[nh + i], hn, y);
        }
        y += __shfl_xor(y, 1, 32);              // pair reduction (wave32)
        if ((t & 1) == 0) ybd[l] = fmaf(ul, Dd, y);
        __syncthreads();
    }
}

// ---------------------------------------------------------------------------
// Launch: setup_inputs order = {L, u, A, Bm, Cm, D, Delta}
// Workspace layout (floats): BuT (b,L,N) | CuT (b,L,N) | deltaT (b,L,d) = 36 MB
// ---------------------------------------------------------------------------
extern "C" void kernel_launch(void* const* d_in, const int* in_sizes, int n_in,
                              void* d_out, int out_size, void* d_ws, size_t ws_size,
                              hipStream_t stream)
{
    (void)in_sizes; (void)n_in; (void)out_size; (void)ws_size;

    const float* u     = (const float*)d_in[1];
    const float* A     = (const float*)d_in[2];
    const float* Bm    = (const float*)d_in[3];
    const float* Cm    = (const float*)d_in[4];
    const float* D     = (const float*)d_in[5];
    const float* Delta = (const float*)d_in[6];
    float* out = (float*)d_out;

    constexpr int batch = 4, dmodel = 1024, N = 64, L = 2048;

    float* BuT = (float*)d_ws;
    float* CuT = BuT + (size_t)batch * L * N;
    float* dT  = CuT + (size_t)batch * L * N;

    const dim3 blk(128, 1, 1);

    // Bu^T = (Bm * u)^T : (b, L, N)
    gemm_f32_wmma<dmodel, L><<<dim3(L / 32, N / 32, batch), blk, 0, stream>>>(
        Bm, u, BuT, N);
    // Cu^T : (b, L, N)
    gemm_f32_wmma<dmodel, L><<<dim3(L / 32, N / 32, batch), blk, 0, stream>>>(
        Cm, u, CuT, N);
    // delta^T = (Delta * u)^T : (b, L, d)
    gemm_f32_wmma<dmodel, L><<<dim3(L / 32, dmodel / 32, batch), blk, 0, stream>>>(
        Delta, u, dT, dmodel);

    // sequential scan + skip connection
    ssm_scan<<<dim3(dmodel / 64, batch), dim3(128, 1, 1), 0, stream>>>(
        u, A, D, BuT, CuT, dT, out, dmodel, N, L);
}